// Gaussian_Layer_37331855736976
// MI455X (gfx1250) — compile-verified
//
#include <hip/hip_runtime.h>
#include <math.h>

#define T_ROWS 100000
#define IN_DIM 768
#define D_G    5
#define N_G    100
#define LN_EPS 1e-5f

#define CK     32                 // K-chunk per LDS buffer
#define XSTR   36                 // padded x row stride (floats): bank-conflict free
#define WSTR   (IN_DIM + 4)       // padded W row stride (floats): bank-conflict free
#define CHUNKS (IN_DIM / CK)      // 24
#define BUF_F  (16 * XSTR)        // floats per x buffer
#define WPB    8                  // waves per block (256 threads, wave32)

typedef __attribute__((ext_vector_type(2))) float v2f;
typedef __attribute__((ext_vector_type(8))) float v8f;

// ---- CDNA5 async global->LDS copy (ASYNCcnt-tracked, no VGPR round trip) ----
__device__ __forceinline__ void async_ld_b128(unsigned lds_off, const float* g) {
    asm volatile("global_load_async_to_lds_b128 %0, %1, off"
                 :: "v"(lds_off), "v"(g) : "memory");
}
__device__ __forceinline__ void wait_ds0()      { asm volatile("s_wait_dscnt 0x0" ::: "memory"); }
__device__ __forceinline__ void wait_async(int) { }
__device__ __forceinline__ void wait_async4()   { asm volatile("s_wait_asynccnt 0x4" ::: "memory"); }
__device__ __forceinline__ void wait_async0()   { asm volatile("s_wait_asynccnt 0x0" ::: "memory"); }

// ---------------------------------------------------------------------------
// Kernel 0: invert the 100 5x5 covariances (Gauss-Jordan, SPD -> no pivoting),
// compute c2[k] = d*log(2*pi) + logdet_k, and zero the global partial sums.
// ---------------------------------------------------------------------------
__global__ void gm_init_kernel(const float* __restrict__ cov,
                               float* __restrict__ invc,   // [N_G*25]
                               float* __restrict__ c2,     // [N_G]
                               float* __restrict__ gpart)  // [N_G]
{
    int k = blockIdx.x * blockDim.x + threadIdx.x;
    if (k < N_G) gpart[k] = 0.f;
    if (k >= N_G) return;

    float a[D_G][2 * D_G];
    for (int i = 0; i < D_G; ++i)
        for (int j = 0; j < D_G; ++j) {
            a[i][j]       = cov[k * 25 + i * 5 + j];
            a[i][D_G + j] = (i == j) ? 1.f : 0.f;
        }
    float logdet = 0.f;
    for (int p = 0; p < D_G; ++p) {
        float piv = a[p][p];
        logdet += logf(piv);
        float ip = 1.f / piv;
        for (int j = 0; j < 2 * D_G; ++j) a[p][j] *= ip;
        for (int r = 0; r < D_G; ++r) {
            if (r == p) continue;
            float f = a[r][p];
            for (int j = 0; j < 2 * D_G; ++j) a[r][j] -= f * a[p][j];
        }
    }
    for (int i = 0; i < D_G; ++i)
        for (int j = 0; j < D_G; ++j)
            invc[k * 25 + i * 5 + j] = a[i][D_G + j];
    c2[k] = (float)D_G * logf(2.f * 3.14159265358979323846f) + logdet;
}

// ---------------------------------------------------------------------------
// Kernel 1: z = sigmoid(x @ W^T + b) via V_WMMA_F32_16X16X4_F32.
//   A = 16-row x tile (M=rows), B = W^T chunk (N=5 valid cols + zero row 5),
//   K loop over 768 in double-buffered, async-staged chunks of 32.
// WMMA fragment layouts (ISA 7.12.2, 32-bit):
//   A 16x4 : lane n -> M = n&15 ; VGPR v -> K = v + 2*(n>=16)
//   B 4x16 : lane n -> N = n&15 ; VGPR v -> K = v + 2*(n>=16)
//   C/D    : lane n -> N = n&15 ; VGPR v -> M = v + 8*(n>=16)
// ---------------------------------------------------------------------------
__global__ __launch_bounds__(256) void gm_gemm_kernel(
    const float* __restrict__ x, const float* __restrict__ W,
    const float* __restrict__ bias, float* __restrict__ zws)
{
    __shared__ float Wlds[(D_G + 1) * WSTR];   // 6 rows, row 5 = zeros (pad N)
    __shared__ float xs[WPB][2][BUF_F];        // wave-private double buffers

    const int tid  = threadIdx.x;
    const int lane = tid & 31;
    const int wave = tid >> 5;

    // stage W (padded stride, zero row 5) once per block
    for (int i = tid; i < D_G * IN_DIM; i += blockDim.x)
        Wlds[(i / IN_DIM) * WSTR + (i % IN_DIM)] = W[i];
    for (int i = tid; i < IN_DIM; i += blockDim.x)
        Wlds[D_G * WSTR + i] = 0.f;
    __syncthreads();

    const int   j  = lane & 15;              // A row index / B col index
    const int   hi = lane >> 4;              // lane-half
    const int   jj = (j < D_G) ? j : D_G;    // zero row for padded B cols
    const float bj = (j < D_G) ? bias[j] : 0.f;

    // per-lane staging coords: 32 lanes x b128 = 4 rows x 32 floats per issue
    const int srow = lane >> 3;              // 0..3
    const int scol = (lane & 7) * 4;         // 0,4,...,28
    unsigned ldsb[2][4];
    #pragma unroll
    for (int bb = 0; bb < 2; ++bb)
        #pragma unroll
        for (int i = 0; i < 4; ++i)
            ldsb[bb][i] = (unsigned)(uintptr_t)&xs[wave][bb][(i * 4 + srow) * XSTR + scol];

    const int numTiles = T_ROWS / 16;        // 6250 (exact)
    const int gwave    = blockIdx.x * WPB + wave;
    const int wstride  = gridDim.x * WPB;

    for (int tile = gwave; tile < numTiles; tile += wstride) {
        const int rowBase = tile * 16;
        const float* gb[4];
        #pragma unroll
        for (int i = 0; i < 4; ++i)
            gb[i] = x + (size_t)(rowBase + i * 4 + srow) * IN_DIM + scol;

        // prologue: chunk 0 -> buffer 0
        wait_ds0();
        #pragma unroll
        for (int i = 0; i < 4; ++i) async_ld_b128(ldsb[0][i], gb[i]);

        v8f acc = {};
        int bb = 0;
        for (int kci = 0; kci < CHUNKS; ++kci) {
            const int kc = kci * CK;
            if (kci + 1 < CHUNKS) {
                // prefetch next chunk into the other buffer, then wait for
                // the current chunk only (async loads complete in order)
                wait_ds0();
                #pragma unroll
                for (int i = 0; i < 4; ++i)
                    async_ld_b128(ldsb[bb ^ 1][i], gb[i] + kc + CK);
                wait_async4();
            } else {
                wait_async0();
            }
            const float* xw = &xs[wave][bb][0];
            #pragma unroll
            for (int kk = 0; kk < CK; kk += 4) {
                v2f a  = *(const v2f*)(xw + j * XSTR + kk + 2 * hi);
                v2f wb = *(const v2f*)(Wlds + jj * WSTR + kc + kk + 2 * hi);
                acc = __builtin_amdgcn_wmma_f32_16x16x4_f32(
                        false, a, false, wb, (short)0, acc, false, false);
            }
            bb ^= 1;
        }
        // epilogue: bias + sigmoid, store z rows (lanes with valid N only)
        if (j < D_G) {
            #pragma unroll
            for (int v = 0; v < 8; ++v) {
                int   row = rowBase + v + 8 * hi;
                float zz  = acc[v] + bj;
                float s   = 1.f / (1.f + __expf(-zz));
                zws[(size_t)row * D_G + j] = s;
            }
        }
    }
}

// ---------------------------------------------------------------------------
// Kernel 2: per-row Gaussian likelihoods + LayerNorm over the 100 gaussians,
// accumulate per-gaussian sums over rows (LDS atomics -> global atomics).
// Two-pass lik computation avoids a 100-element per-thread array.
// ---------------------------------------------------------------------------
__global__ __launch_bounds__(256) void gm_gauss_kernel(
    const float* __restrict__ zws, const float* __restrict__ centers,
    const float* __restrict__ invc, const float* __restrict__ c2,
    const float* __restrict__ gamma, const float* __restrict__ beta,
    float* __restrict__ gpart)
{
    __shared__ float sacc[N_G];
    __shared__ float sc[N_G * D_G];
    __shared__ float sinv[N_G * 25];
    __shared__ float sc2[N_G];
    __shared__ float sg[N_G];
    __shared__ float sb[N_G];

    for (int i = threadIdx.x; i < N_G; i += blockDim.x) {
        sacc[i] = 0.f; sc2[i] = c2[i]; sg[i] = gamma[i]; sb[i] = beta[i];
    }
    for (int i = threadIdx.x; i < N_G * D_G; i += blockDim.x) sc[i] = centers[i];
    for (int i = threadIdx.x; i < N_G * 25;  i += blockDim.x) sinv[i] = invc[i];
    __syncthreads();

    const int t = blockIdx.x * blockDim.x + threadIdx.x;
    if (t < T_ROWS) {
        float z[D_G];
        #pragma unroll
        for (int i = 0; i < D_G; ++i) z[i] = zws[(size_t)t * D_G + i];

        float sum = 0.f, sumsq = 0.f;
        for (int k = 0; k < N_G; ++k) {
            const float* c = sc + k * D_G;
            const float* m = sinv + k * 25;
            float d[D_G];
            #pragma unroll
            for (int i = 0; i < D_G; ++i) d[i] = z[i] - c[i];
            float maha = 0.f;
            #pragma unroll
            for (int i = 0; i < D_G; ++i) {
                float r = 0.f;
                #pragma unroll
                for (int jx = 0; jx < D_G; ++jx) r += m[i * 5 + jx] * d[jx];
                maha += d[i] * r;
            }
            float lik = __expf(-0.5f * (maha + sc2[k]));
            sum += lik; sumsq += lik * lik;
        }
        const float mu   = sum * (1.f / N_G);
        const float var  = sumsq * (1.f / N_G) - mu * mu;
        const float istd = rsqrtf(var + LN_EPS);

        for (int k = 0; k < N_G; ++k) {
            const float* c = sc + k * D_G;
            const float* m = sinv + k * 25;
            float d[D_G];
            #pragma unroll
            for (int i = 0; i < D_G; ++i) d[i] = z[i] - c[i];
            float maha = 0.f;
            #pragma unroll
            for (int i = 0; i < D_G; ++i) {
                float r = 0.f;
                #pragma unroll
                for (int jx = 0; jx < D_G; ++jx) r += m[i * 5 + jx] * d[jx];
                maha += d[i] * r;
            }
            float lik    = __expf(-0.5f * (maha + sc2[k]));
            float normed = (lik - mu) * istd * sg[k] + sb[k];
            atomicAdd(&sacc[k], normed);
        }
    }
    __syncthreads();
    for (int i = threadIdx.x; i < N_G; i += blockDim.x)
        atomicAdd(&gpart[i], sacc[i]);
}

// ---------------------------------------------------------------------------
// Kernel 3: out[k] = gpart[k] / T
// ---------------------------------------------------------------------------
__global__ void gm_final_kernel(const float* __restrict__ gpart,
                                float* __restrict__ out)
{
    int k = threadIdx.x;
    if (k < N_G) out[k] = gpart[k] * (1.f / (float)T_ROWS);
}

// ---------------------------------------------------------------------------
extern "C" void kernel_launch(void* const* d_in, const int* in_sizes, int n_in,
                              void* d_out, int out_size, void* d_ws, size_t ws_size,
                              hipStream_t stream) {
    const float* x       = (const float*)d_in[0];
    const float* W       = (const float*)d_in[1];
    const float* b       = (const float*)d_in[2];
    const float* centers = (const float*)d_in[3];
    const float* cov     = (const float*)d_in[4];
    const float* gamma   = (const float*)d_in[5];
    const float* beta    = (const float*)d_in[6];
    float* out = (float*)d_out;

    // workspace layout
    float* zws   = (float*)d_ws;                     // T_ROWS * D_G
    float* invc  = zws + (size_t)T_ROWS * D_G;       // N_G * 25
    float* c2    = invc + N_G * 25;                  // N_G
    float* gpart = c2 + N_G;                         // N_G

    gm_init_kernel<<<1, 128, 0, stream>>>(cov, invc, c2, gpart);

    const int numTiles = T_ROWS / 16;                // 6250
    const int blocksA  = (numTiles + WPB - 1) / WPB; // one tile per wave
    gm_gemm_kernel<<<blocksA, 256, 0, stream>>>(x, W, b, zws);

    const int blocksB = (T_ROWS + 255) / 256;
    gm_gauss_kernel<<<blocksB, 256, 0, stream>>>(zws, centers, invc, c2,
                                                 gamma, beta, gpart);

    gm_final_kernel<<<1, 128, 0, stream>>>(gpart, out);
}